// GCN_10110353015121
// MI455X (gfx1250) — compile-verified
//
#include <hip/hip_runtime.h>
#include <math.h>

typedef float v2f __attribute__((ext_vector_type(2)));
typedef float v8f __attribute__((ext_vector_type(8)));

#define N_NODES 20000
#define N_EDGES 256000
#define EMB 300
#define EMB_P 304          // EMB padded up to a multiple of 16 (for ragged N tiles)
#define FEAT 256
#define N_LAYERS 5
#define N_GRAPHS 256
#define BN_EPS 1e-5f

static inline int cdiv(int a, int b) { return (a + b - 1) / b; }

// ---------------- node embedding: h = atom_emb[x_atom] + chir_emb[x_chir] ----------------
__global__ void k_embed(const int* __restrict__ xa, const int* __restrict__ xc,
                        const float* __restrict__ aemb, const float* __restrict__ cemb,
                        float* __restrict__ h) {
  int idx = blockIdx.x * blockDim.x + threadIdx.x;
  if (idx >= N_NODES * EMB) return;
  int i = idx / EMB, f = idx - i * EMB;
  h[idx] = aemb[xa[i] * EMB + f] + cemb[xc[i] * EMB + f];
}

__global__ void k_zero(float* __restrict__ p, int n) {
  int idx = blockIdx.x * blockDim.x + threadIdx.x;
  if (idx < n) p[idx] = 0.0f;
}

// Build padded+transposed GCN weights: Wt[l][n][k] = W[l][k][n], n in [0,304), zero for n>=300.
__global__ void k_prep_gcn_wt(const float* __restrict__ W, float* __restrict__ Wt) {
  int idx = blockIdx.x * blockDim.x + threadIdx.x;
  if (idx >= N_LAYERS * EMB_P * EMB) return;
  int l = idx / (EMB_P * EMB);
  int r = idx - l * (EMB_P * EMB);
  int n = r / EMB, k = r - n * EMB;
  Wt[idx] = (n < EMB) ? W[(size_t)l * EMB * EMB + k * EMB + n] : 0.0f;
}

// Transposed feat_w: Ft[n][k] = feat_w[k][n]  (256 x 300)
__global__ void k_prep_feat_wt(const float* __restrict__ W, float* __restrict__ Wt) {
  int idx = blockIdx.x * blockDim.x + threadIdx.x;
  if (idx >= FEAT * EMB) return;
  int n = idx / EMB, k = idx - n * EMB;
  Wt[idx] = W[k * FEAT + n];
}

// ---------------- f32 WMMA GEMM: C[M x Ncols] = A[M x K] * Bt^T (+bias) ------------------
// Bt is the TRANSPOSED B: Bt[n][k], row stride ldbt, with n padded so every lane's load is
// in bounds -> no guards inside the K loop, A and B frags each load as one b64.
// One wave32 per 16x16 C tile via V_WMMA_F32_16X16X4_F32. M%16==0, K%4==0.
__global__ void k_wmma_gemm_f32(const float* __restrict__ A, int lda,
                                const float* __restrict__ Bt, int ldbt,
                                float* __restrict__ C, int ldc,
                                const float* __restrict__ bias,
                                int M, int K, int Ncols, int tilesN) {
  const int lane = threadIdx.x & 31;
  const int wave = (blockIdx.x * blockDim.x + threadIdx.x) >> 5;
  const int tilesM = M >> 4;
  if (wave >= tilesM * tilesN) return;        // wave-uniform exit; EXEC full for WMMA
  const int tm = wave / tilesN;
  const int tn = wave - tm * tilesN;
  const int m0 = tm << 4, n0 = tn << 4;
  const int half = lane >> 4;                 // lane-half owns K pair (k,k+1) or (k+2,k+3)
  const int l16  = lane & 15;
  const int arow = m0 + l16;                  // M index of this lane's A row
  const int bcol = n0 + l16;                  // N index of this lane's B column / C column

  const float* ap = A  + (size_t)arow * lda  + (half << 1);
  const float* bp = Bt + (size_t)bcol * ldbt + (half << 1);

  v8f c = {};
#pragma unroll 5
  for (int k = 0; k < K; k += 4) {
    v2f a = *(const v2f*)ap;                  // A 16x4 frag (contiguous pair -> b64)
    v2f b = *(const v2f*)bp;                  // B 4x16 frag via transposed copy -> b64
    ap += 4; bp += 4;
    c = __builtin_amdgcn_wmma_f32_16x16x4_f32(false, a, false, b, (short)0, c,
                                              false, false);
  }
  if (bcol < Ncols) {                         // one-time ragged-column store guard
    const float bv = bias ? bias[bcol] : 0.0f;
    const int rbase = m0 + half * 8;          // C/D layout: lane-half selects M or M+8
#pragma unroll
    for (int r = 0; r < 8; ++r)
      C[(size_t)(rbase + r) * ldc + bcol] = c[r] + bv;
  }
}

// ---------------- edge gather + per-edge scalar + atomic scatter (segment_sum) -----------
__global__ void k_scatter(const float* __restrict__ hW,
                          const int* __restrict__ ei,      // [2*E]: [0,E)=row(src), [E,2E)=col(dst)
                          const int* __restrict__ etype,
                          const int* __restrict__ edir,
                          const float* __restrict__ e1,    // [6] this layer
                          const float* __restrict__ e2,    // [3] this layer
                          float* __restrict__ hout) {
  const int e = blockIdx.x;
  int row, col, t, d;
  if (e < N_EDGES) {
    row = ei[e]; col = ei[N_EDGES + e]; t = etype[e]; d = edir[e];
  } else {                                    // appended self loops: etype=5, edir=0
    const int i = e - N_EDGES;
    row = i; col = i; t = 5; d = 0;
  }
  const float ee = e1[t] + e2[d];
  const int f = threadIdx.x;
  if (f < EMB) {
    const float m = hW[row * EMB + f] + ee;
    unsafeAtomicAdd(&hout[col * EMB + f], m); // native global_atomic_add_f32
  }
}

// ---------------- BatchNorm stats: one block per column, biased variance -----------------
__global__ void k_bn_stats(const float* __restrict__ h,
                           float* __restrict__ mean, float* __restrict__ rstd) {
  const int c = blockIdx.x;                   // 0..EMB-1
  float s = 0.0f, s2 = 0.0f;
  for (int i = threadIdx.x; i < N_NODES; i += blockDim.x) {
    const float v = h[i * EMB + c];
    s += v; s2 += v * v;
  }
  __shared__ float sh[256], sh2[256];
  sh[threadIdx.x] = s; sh2[threadIdx.x] = s2;
  __syncthreads();
  for (int off = blockDim.x >> 1; off > 0; off >>= 1) {
    if (threadIdx.x < (unsigned)off) {
      sh[threadIdx.x]  += sh[threadIdx.x + off];
      sh2[threadIdx.x] += sh2[threadIdx.x + off];
    }
    __syncthreads();
  }
  if (threadIdx.x == 0) {
    const float m = sh[0] / (float)N_NODES;
    const float var = sh2[0] / (float)N_NODES - m * m;
    mean[c] = m;
    rstd[c] = rsqrtf(var + BN_EPS);
  }
}

// (gcn_bias cancels: BN subtracts the per-column mean, which absorbs per-column constants)
__global__ void k_bn_apply(float* __restrict__ h,
                           const float* __restrict__ mean, const float* __restrict__ rstd,
                           const float* __restrict__ gamma, const float* __restrict__ beta,
                           int relu) {
  int idx = blockIdx.x * blockDim.x + threadIdx.x;
  if (idx >= N_NODES * EMB) return;
  int c = idx % EMB;
  float v = (h[idx] - mean[c]) * rstd[c] * gamma[c] + beta[c];
  if (relu) v = fmaxf(v, 0.0f);
  h[idx] = v;
}

// ---------------- global mean pool over graphs -------------------------------------------
__global__ void k_pool_sum(const float* __restrict__ h, const int* __restrict__ batch,
                           float* __restrict__ gsum, float* __restrict__ gcnt) {
  int idx = blockIdx.x * blockDim.x + threadIdx.x;
  if (idx >= N_NODES * EMB) return;
  int i = idx / EMB, f = idx - i * EMB;
  int g = batch[i];
  unsafeAtomicAdd(&gsum[g * EMB + f], h[idx]);
  if (f == 0) unsafeAtomicAdd(&gcnt[g], 1.0f);
}

__global__ void k_pool_div(const float* __restrict__ gsum, const float* __restrict__ gcnt,
                           float* __restrict__ hg) {
  int idx = blockIdx.x * blockDim.x + threadIdx.x;
  if (idx >= N_GRAPHS * EMB) return;
  int g = idx / EMB;
  hg[idx] = gsum[idx] / fmaxf(gcnt[g], 1.0f);
}

// ---------------- head: z = softplus(hf @ w1 + b1); pred = z @ w2 + b2 -------------------
__global__ void k_head(const float* __restrict__ hf,
                       const float* __restrict__ w1, const float* __restrict__ b1,
                       const float* __restrict__ w2, const float* __restrict__ b2,
                       float* __restrict__ pred) {
  const int g = blockIdx.x;        // graph
  const int j = threadIdx.x;       // 0..127
  __shared__ float z[FEAT / 2];
  float acc = b1[j];
  for (int k = 0; k < FEAT; ++k) acc += hf[g * FEAT + k] * w1[k * (FEAT / 2) + j];
  z[j] = (acc > 20.0f) ? acc : log1pf(expf(acc));  // softplus
  __syncthreads();
  if (j < 2) {
    float p = b2[j];
    for (int k = 0; k < FEAT / 2; ++k) p += z[k] * w2[k * 2 + j];
    pred[g * 2 + j] = p;
  }
}

extern "C" void kernel_launch(void* const* d_in, const int* in_sizes, int n_in,
                              void* d_out, int out_size, void* d_ws, size_t ws_size,
                              hipStream_t stream) {
  const int*   x_atom   = (const int*)d_in[0];
  const int*   x_chir   = (const int*)d_in[1];
  const int*   edge_idx = (const int*)d_in[2];
  const int*   etype    = (const int*)d_in[3];
  const int*   edir     = (const int*)d_in[4];
  const int*   batch    = (const int*)d_in[5];
  const float* atom_emb = (const float*)d_in[6];
  const float* chir_emb = (const float*)d_in[7];
  const float* gcn_w    = (const float*)d_in[8];
  /* d_in[9] = gcn_bias: cancels inside BatchNorm */
  const float* ee1      = (const float*)d_in[10];  // [5][6]
  const float* ee2      = (const float*)d_in[11];  // [5][3]
  const float* gamma    = (const float*)d_in[12];  // [5][300]
  const float* beta     = (const float*)d_in[13];
  const float* feat_w   = (const float*)d_in[14];  // [300][256]
  const float* feat_b   = (const float*)d_in[15];
  const float* hw1      = (const float*)d_in[16];  // [256][128]
  const float* hb1      = (const float*)d_in[17];
  const float* hw2      = (const float*)d_in[18];  // [128][2]
  const float* hb2      = (const float*)d_in[19];
  float* out = (float*)d_out;                      // hf [256*256] ++ pred [256*2]

  float* ws   = (float*)d_ws;
  float* h    = ws;                          // 6,000,000 f32 (24 MB)
  float* hW   = ws + 6000000;                // 6,000,000 f32 (24 MB)
  float* gwt  = ws + 12000000;               // 5*304*300 = 456,000 (padded+transposed W)
  float* fwt  = gwt + N_LAYERS * EMB_P * EMB;// 256*300  =  76,800 (transposed feat_w)
  float* gsum = fwt + FEAT * EMB;            // 76,800
  float* gcnt = gsum + N_GRAPHS * EMB;       // 256 (contiguous with gsum for one zero pass)
  float* hg   = gcnt + N_GRAPHS;             // 76,800
  float* mean = hg + N_GRAPHS * EMB;         // 300
  float* rstd = mean + EMB;                  // 300

  const int NE = N_NODES * EMB;

  // prepare padded/transposed weights once per call
  k_prep_gcn_wt<<<cdiv(N_LAYERS * EMB_P * EMB, 256), 256, 0, stream>>>(gcn_w, gwt);
  k_prep_feat_wt<<<cdiv(FEAT * EMB, 256), 256, 0, stream>>>(feat_w, fwt);

  // node embedding
  k_embed<<<cdiv(NE, 256), 256, 0, stream>>>(x_atom, x_chir, atom_emb, chir_emb, h);

  // 5 GCN layers
  const int tilesN = cdiv(EMB, 16);                     // 19 (last tile ragged, store-guarded)
  const int waves1 = (N_NODES / 16) * tilesN;           // 23750 tiles
  for (int l = 0; l < N_LAYERS; ++l) {
    k_wmma_gemm_f32<<<cdiv(waves1, 4), 128, 0, stream>>>(
        h, EMB, gwt + (size_t)l * EMB_P * EMB, EMB, hW, EMB, nullptr,
        N_NODES, EMB, EMB, tilesN);
    k_zero<<<cdiv(NE, 256), 256, 0, stream>>>(h, NE);
    k_scatter<<<N_EDGES + N_NODES, 320, 0, stream>>>(
        hW, edge_idx, etype, edir, ee1 + l * 6, ee2 + l * 3, h);
    k_bn_stats<<<EMB, 256, 0, stream>>>(h, mean, rstd);
    k_bn_apply<<<cdiv(NE, 256), 256, 0, stream>>>(
        h, mean, rstd, gamma + l * EMB, beta + l * EMB, (l < N_LAYERS - 1) ? 1 : 0);
  }

  // global mean pool
  k_zero<<<cdiv(N_GRAPHS * EMB + N_GRAPHS, 256), 256, 0, stream>>>(gsum, N_GRAPHS * EMB + N_GRAPHS);
  k_pool_sum<<<cdiv(NE, 256), 256, 0, stream>>>(h, batch, gsum, gcnt);
  k_pool_div<<<cdiv(N_GRAPHS * EMB, 256), 256, 0, stream>>>(gsum, gcnt, hg);

  // hf = hg @ feat_w + feat_b  -> first 65536 floats of d_out (WMMA, exact 256 cols)
  const int waves2 = (N_GRAPHS / 16) * (FEAT / 16);     // 256 tiles
  k_wmma_gemm_f32<<<cdiv(waves2, 4), 128, 0, stream>>>(
      hg, EMB, fwt, EMB, out, FEAT, feat_b,
      N_GRAPHS, EMB, FEAT, FEAT / 16);

  // pred -> last 512 floats of d_out
  k_head<<<N_GRAPHS, FEAT / 2, 0, stream>>>(out, hw1, hb1, hw2, hb2, out + N_GRAPHS * FEAT);
}